// quantum_hidden_Markov_neural_network_61366492725511
// MI455X (gfx1250) — compile-verified
//
#include <hip/hip_runtime.h>
#include <hip/hip_bf16.h>

typedef __attribute__((ext_vector_type(2))) float v2f;
typedef __attribute__((ext_vector_type(8))) float v8f;

// ---------------------------------------------------------------------------
// Prep kernel (1 thread, trivial work): fold 5 layers x 5 Kraus ops plus the
// 2 measurement ops into one real 8x16 weight matrix W, and store it in the
// WMMA B-operand lane layout (B 4x16 f32: element(K = vgpr + 2*laneHalf,
// N = lane%16)).  ws layout: [B1.v0 | B1.v1 | B2.v0 | B2.v1], 32 floats each.
// ---------------------------------------------------------------------------
__global__ void qhmm_prep(const float* __restrict__ kre, const float* __restrict__ kim,
                          const float* __restrict__ mre, const float* __restrict__ mim,
                          float* __restrict__ ws) {
    if (threadIdx.x != 0 || blockIdx.x != 0) return;

    // Composite channel superoperator S (4x4 complex), S = S5 @ ... @ S1
    float Sre[4][4], Sim[4][4];
    for (int i = 0; i < 4; ++i)
        for (int j = 0; j < 4; ++j) { Sre[i][j] = (i == j) ? 1.f : 0.f; Sim[i][j] = 0.f; }

    for (int l = 0; l < 5; ++l) {
        float Lre[4][4] = {}, Lim[4][4] = {};
        for (int k = 0; k < 5; ++k) {
            const float* kr = kre + (l * 5 + k) * 4;
            const float* ki = kim + (l * 5 + k) * 4;
            for (int a = 0; a < 2; ++a)
            for (int d = 0; d < 2; ++d)
            for (int b = 0; b < 2; ++b)
            for (int c = 0; c < 2; ++c) {
                const int j = a * 2 + d, p = b * 2 + c;
                const float rAB = kr[a * 2 + b], iAB = ki[a * 2 + b];
                const float rDC = kr[d * 2 + c], iDC = ki[d * 2 + c];
                // K[a][b] * conj(K[d][c])
                Lre[j][p] += rAB * rDC + iAB * iDC;
                Lim[j][p] += iAB * rDC - rAB * iDC;
            }
        }
        float Nre[4][4], Nim[4][4];
        for (int i = 0; i < 4; ++i)
        for (int j = 0; j < 4; ++j) {
            float ar = 0.f, ai = 0.f;
            for (int t = 0; t < 4; ++t) {
                ar += Lre[i][t] * Sre[t][j] - Lim[i][t] * Sim[t][j];
                ai += Lre[i][t] * Sim[t][j] + Lim[i][t] * Sre[t][j];
            }
            Nre[i][j] = ar; Nim[i][j] = ai;
        }
        for (int i = 0; i < 4; ++i)
            for (int j = 0; j < 4; ++j) { Sre[i][j] = Nre[i][j]; Sim[i][j] = Nim[i][j]; }
    }

    // W[8 real-in rows][16 real-out cols]; col = m*8 + j*2 + (0:re,1:im)
    float W[8][16];
    for (int m = 0; m < 2; ++m) {
        float Tre[4][4] = {}, Tim[4][4] = {};
        const float* mr = mre + m * 4;
        const float* mi = mim + m * 4;
        for (int a = 0; a < 2; ++a)
        for (int d = 0; d < 2; ++d)
        for (int b = 0; b < 2; ++b)
        for (int c = 0; c < 2; ++c) {
            const int j = a * 2 + d, p = b * 2 + c;
            const float rAB = mr[a * 2 + b], iAB = mi[a * 2 + b];
            const float rDC = mr[d * 2 + c], iDC = mi[d * 2 + c];
            Tre[j][p] = rAB * rDC + iAB * iDC;
            Tim[j][p] = iAB * rDC - rAB * iDC;
        }
        for (int j = 0; j < 4; ++j)
        for (int p = 0; p < 4; ++p) {
            float ar = 0.f, ai = 0.f;
            for (int t = 0; t < 4; ++t) {
                ar += Tre[j][t] * Sre[t][p] - Tim[j][t] * Sim[t][p];
                ai += Tre[j][t] * Sim[t][p] + Tim[j][t] * Sre[t][p];
            }
            const int col = m * 8 + j * 2;
            W[p][col]         = ar;   // re <- re part
            W[4 + p][col]     = -ai;  // re <- im part
            W[p][col + 1]     = ai;   // im <- re part
            W[4 + p][col + 1] = ar;   // im <- im part
        }
    }

    // Emit in WMMA-B lane order: K = r + 2*h, N = n.
    for (int r = 0; r < 2; ++r)
    for (int h = 0; h < 2; ++h)
    for (int n = 0; n < 16; ++n) {
        ws[r * 32 + h * 16 + n]      = W[r + 2 * h][n];      // B1: K rows 0..3 (re inputs)
        ws[64 + r * 32 + h * 16 + n] = W[4 + r + 2 * h][n];  // B2: K rows 4..7 (im inputs)
    }
}

// ---------------------------------------------------------------------------
// Streaming kernel: each wave processes tiles of 16 batch elements.
// A-operand (16x4 f32): lane L<16 -> (M=L, K=0,1), lane L+16 -> (M=L, K=2,3)
//   == one float2 load from the row-major [B,2,2] array.
// D (16x16 f32): vgpr r, lane -> (M = r + 8*(lane>>4), N = lane&15);
//   N = m*8 + j*2 + c maps directly onto interleaved-complex out[m][n][j].
// ---------------------------------------------------------------------------
__global__ void __launch_bounds__(256)
qhmm_main(const float* __restrict__ xre, const float* __restrict__ xim,
          const float* __restrict__ ws, float* __restrict__ out,
          int ntiles, long long B8) {
    const int lane = threadIdx.x & 31;
    const int half = lane >> 4;        // 0/1 : M offset 0 / 8, K offset 0 / 2
    const int lm   = lane & 15;        // M (loads) / N (stores)
    const int msel = (lane >> 3) & 1;  // output class from column
    const int q    = lane & 7;         // float offset within 4-complex chunk

    v2f b1, b2;                        // uniform B operands, loaded once
    b1.x = ws[lane];       b1.y = ws[32 + lane];
    b2.x = ws[64 + lane];  b2.y = ws[96 + lane];

    const int wave   = blockIdx.x * (blockDim.x >> 5) + (threadIdx.x >> 5);
    const int nwaves = gridDim.x * (blockDim.x >> 5);

    // per-lane invariant pieces of the addresses
    const long long ldoff = (long long)lm * 4 + half * 2;              // floats
    float* const obase    = out + (long long)msel * B8 + (long long)(8 * half) * 8 + q;

    for (int t = wave; t < ntiles; t += nwaves) {
        const long long n0 = (long long)t * 16;

        v2f a1 = __builtin_nontemporal_load((const v2f*)(xre + n0 * 4 + ldoff));
        v2f a2 = __builtin_nontemporal_load((const v2f*)(xim + n0 * 4 + ldoff));

        v8f c = {};
        // D = X_re(16x4) * W_re_rows(4x16)
        c = __builtin_amdgcn_wmma_f32_16x16x4_f32(false, a1, false, b1,
                                                  (short)0, c, false, false);
        // D += X_im(16x4) * W_im_rows(4x16)
        c = __builtin_amdgcn_wmma_f32_16x16x4_f32(false, a2, false, b2,
                                                  (short)0, c, false, false);

        float* o = obase + n0 * 8;
#pragma unroll
        for (int r = 0; r < 8; ++r)
            __builtin_nontemporal_store(c[r], o + (long long)r * 8);
    }
}

extern "C" void kernel_launch(void* const* d_in, const int* in_sizes, int n_in,
                              void* d_out, int out_size, void* d_ws, size_t ws_size,
                              hipStream_t stream) {
    const float* xre = (const float*)d_in[0];  // [B,2,2]
    const float* xim = (const float*)d_in[1];  // [B,2,2]
    const float* kre = (const float*)d_in[2];  // [5,5,2,2]
    const float* kim = (const float*)d_in[3];
    const float* mre = (const float*)d_in[4];  // [2,2,2]
    const float* mim = (const float*)d_in[5];
    float* out = (float*)d_out;                // [2,B,2,2] complex64 -> float pairs
    float* ws  = (float*)d_ws;                 // needs 512 B

    const long long B = (long long)in_sizes[0] / 4;
    const int ntiles  = (int)(B / 16);

    qhmm_prep<<<1, 1, 0, stream>>>(kre, kim, mre, mim, ws);

    const int block  = 256;                    // 8 waves (wave32)
    const int blocks = 2048;                   // grid-stride over 262144 tiles
    qhmm_main<<<blocks, block, 0, stream>>>(xre, xim, ws, out, ntiles, B * 8);
}